// CrossEntropyLossHung_67018669687337
// MI455X (gfx1250) — compile-verified
//
#include <hip/hip_runtime.h>
#include <stdint.h>

#define NMAX 64

typedef float v2f __attribute__((ext_vector_type(2)));
typedef float v8f __attribute__((ext_vector_type(8)));
typedef int   v4i __attribute__((ext_vector_type(4)));

#if __has_builtin(__builtin_amdgcn_global_load_async_to_lds_b128)
#define USE_ASYNC_LDS 1
typedef __attribute__((address_space(1))) v4i g_v4i;   // global int4
typedef __attribute__((address_space(3))) v4i l_v4i;   // LDS int4
#else
#define USE_ASYNC_LDS 0
#endif

#if __has_builtin(__builtin_amdgcn_wmma_f32_16x16x4_f32)
#define USE_WMMA_F32X4 1
#else
#define USE_WMMA_F32X4 0
#endif

template <int N>
static __device__ __forceinline__ void wait_asynccnt_le() {
#if USE_ASYNC_LDS
#if __has_builtin(__builtin_amdgcn_s_wait_asynccnt)
  __builtin_amdgcn_s_wait_asynccnt(N);       // N is a constant expression here
#else
  if (N == 0)
    asm volatile("s_wait_asynccnt 0x0" ::: "memory");
  else
    asm volatile("s_wait_asynccnt 0x20" ::: "memory");
#endif
#endif
}

// ---- order-preserving double <-> uint64 key (for wave argmin) ----
static __device__ __forceinline__ unsigned long long okey(double v) {
  unsigned long long u = (unsigned long long)__double_as_longlong(v);
  return (u & 0x8000000000000000ULL) ? ~u : (u | 0x8000000000000000ULL);
}
static __device__ __forceinline__ double okey_inv(unsigned long long k) {
  unsigned long long u =
      (k & 0x8000000000000000ULL) ? (k & 0x7FFFFFFFFFFFFFFFULL) : ~k;
  return __longlong_as_double((long long)u);
}

// fetch a per-lane 2-slot register array element by column index (wave32)
static __device__ __forceinline__ int fetch2i(int a, int b, int col) {
  int x = __shfl(a, col & 31, 32);
  int y = __shfl(b, col & 31, 32);
  return (col >= 32) ? y : x;
}

// BCE element per reference: ali = dis+gt (2 -> 0.9); p = ali*pred; g = ali*gt
static __device__ __forceinline__ float bce_elem(float pr, float g, int d) {
  float ali = (float)d + g;
  if (ali > 1.0f) ali = 0.9f;
  if (ali == 0.0f) return 0.0f;            // g=0, p=0 -> bce = 0 (clamped form)
  float p  = ali * pr;
  float gg = ali * g;
  float l1 = fmaxf(logf(p),     -100.0f);
  float l2 = fmaxf(log1pf(-p),  -100.0f);
  return -(gg * l1 + (1.0f - gg) * l2);
}

__global__ __launch_bounds__(32)
void hung_loss_kernel(const float* __restrict__ pred,
                      const float* __restrict__ gt,
                      const int*   __restrict__ ns,
                      float* __restrict__ partial,
                      int B)
{
  __shared__ float  tile [NMAX * NMAX];       // pred tile (16KB)
  __shared__ float  gtile[NMAX * NMAX];       // gt tile   (16KB)
  __shared__ double uarr [NMAX];              // row duals u[] (f64)

  const int lane = threadIdx.x & 31;
  const int b    = blockIdx.x;
  if (b >= B) return;

  const int n = ns[b];
  const float* __restrict__ srcp = pred + (size_t)b * (NMAX * NMAX);
  const float* __restrict__ gptr = gt   + (size_t)b * (NMAX * NMAX);

  // ---- stage pred tile (32 async b128 ops), then gt tile (32 more).
  //      Async loads complete in order, so ASYNCcnt<=32 == "pred ready";
  //      the gt fetch overlaps the whole Hungarian solve.
  for (int it = 0; it < 32; ++it) {
    const int f4 = it * 32 + lane;            // float4 index, 512B/instruction
#if USE_ASYNC_LDS
    __builtin_amdgcn_global_load_async_to_lds_b128(
        (g_v4i*)(srcp + f4 * 4), (l_v4i*)(&tile[0] + f4 * 4), 0, 0);
#else
    reinterpret_cast<float4*>(&tile[0])[f4] =
        reinterpret_cast<const float4*>(srcp)[f4];
#endif
  }
  for (int it = 0; it < 32; ++it) {
    const int f4 = it * 32 + lane;
#if USE_ASYNC_LDS
    __builtin_amdgcn_global_load_async_to_lds_b128(
        (g_v4i*)(gptr + f4 * 4), (l_v4i*)(&gtile[0] + f4 * 4), 0, 0);
#else
    reinterpret_cast<float4*>(&gtile[0])[f4] =
        reinterpret_cast<const float4*>(gptr)[f4];
#endif
  }
  wait_asynccnt_le<32>();                     // pred tile resident

  // ---- Jonker-Volgenant LAP, minimizing -pred on the n x n submatrix ----
  // Lane owns columns c0=lane, c1=lane+32. Column state in registers.
  uarr[lane]      = 0.0;
  uarr[lane + 32] = 0.0;

  int    p0 = -1,  p1 = -1;                   // row matched to my columns
  double v0 = 0.0, v1 = 0.0;                  // column duals
  const int  c0 = lane, c1 = lane + 32;
  const bool valid0 = (c0 < n), valid1 = (c1 < n);

  for (int i = 0; i < n; ++i) {               // insert row i
    double minv0 = 1e18, minv1 = 1e18;
    int    way0  = NMAX, way1  = NMAX;
    bool   used0 = !valid0, used1 = !valid1;  // invalid cols permanently used
    int    j0 = NMAX;                         // virtual start column
    int    i0 = i;                            // row matched to j0
    int    freecol = -1;

    for (int guard = 0; guard <= NMAX; ++guard) {
      const double ui0 = uarr[i0];
      if (!used0) {
        double cur = -(double)tile[i0 * NMAX + c0] - ui0 - v0;
        if (cur < minv0) { minv0 = cur; way0 = j0; }
      }
      if (!used1) {
        double cur = -(double)tile[i0 * NMAX + c1] - ui0 - v1;
        if (cur < minv1) { minv1 = cur; way1 = j0; }
      }
      // wave argmin over free columns: min ordered key, tie -> smallest col
      unsigned long long k0 = used0 ? ~0ULL : okey(minv0);
      unsigned long long k1 = used1 ? ~0ULL : okey(minv1);
      unsigned long long kb = (k1 < k0) ? k1 : k0;
      for (int off = 16; off >= 1; off >>= 1) {
        unsigned long long o = __shfl_xor(kb, off, 32);
        if (o < kb) kb = o;
      }
      const double delta = okey_inv(kb);
      int cc = 127;
      if (k0 == kb) cc = c0;
      if (k1 == kb && c1 < cc) cc = c1;
      for (int off = 16; off >= 1; off >>= 1) {
        int o = __shfl_xor(cc, off, 32);
        if (o < cc) cc = o;
      }
      const int j1 = cc;

      // dual updates over used set (distinct target rows -> race-free DS RMW)
      if (lane == 0) uarr[i] += delta;        // virtual column -> u[i]
      if (used0 && valid0) { uarr[p0] += delta; v0 -= delta; }
      if (used1 && valid1) { uarr[p1] += delta; v1 -= delta; }
      if (!used0) minv0 -= delta;
      if (!used1) minv1 -= delta;

      j0 = j1;
      const int pj = fetch2i(p0, p1, j1);
      if (pj < 0) { freecol = j1; break; }    // reached a free column
      i0 = pj;
      if (lane == (j1 & 31)) {                // mark j1 used (owner lane)
        if (j1 < 32) used0 = true; else used1 = true;
      }
    }

    // augment: walk the alternating tree back to the virtual column
    int jc = freecol;
    while (jc >= 0 && jc != NMAX) {
      const int jp = fetch2i(way0, way1, jc);
      const int np = (jp == NMAX) ? i : fetch2i(p0, p1, jp);
      if (lane == (jc & 31)) {
        if (jc < 32) p0 = np; else p1 = np;
      }
      jc = jp;
    }
  }

  wait_asynccnt_le<0>();                      // gt tile resident

  // ---- loss: BCE, reduced on the matrix pipe (V_WMMA_F32_16X16X4_F32) ----
#if USE_WMMA_F32X4
  v8f acc = {0.f, 0.f, 0.f, 0.f, 0.f, 0.f, 0.f, 0.f};
  const v2f ones = {1.0f, 1.0f};
#else
  float ssum = 0.0f;
#endif
  for (int r = 0; r < n; ++r) {
    float b0 = 0.0f, b1 = 0.0f;
    if (valid0) b0 = bce_elem(tile[r * NMAX + c0], gtile[r * NMAX + c0],
                              (p0 == r) ? 1 : 0);
    if (valid1) b1 = bce_elem(tile[r * NMAX + c1], gtile[r * NMAX + c1],
                              (p1 == r) ? 1 : 0);
#if USE_WMMA_F32X4
    // A (16x4 f32, 2 VGPRs) = 64 bce values, B = ones: D accumulates row sums
    // replicated over 16 columns -> grand total = (sum of acc)/16, exact f32.
    v2f av = {b0, b1};
    acc = __builtin_amdgcn_wmma_f32_16x16x4_f32(
        false, av, false, ones, (short)0, acc, false, false);
#else
    ssum += b0 + b1;
#endif
  }
#if USE_WMMA_F32X4
  float s = acc[0] + acc[1] + acc[2] + acc[3] +
            acc[4] + acc[5] + acc[6] + acc[7];
  s *= (1.0f / 16.0f);
#else
  float s = ssum;
#endif
  for (int off = 16; off >= 1; off >>= 1) s += __shfl_xor(s, off, 32);
  if (lane == 0) partial[b] = s;
}

__global__ __launch_bounds__(256)
void hung_finalize_kernel(const float* __restrict__ partial,
                          const int*   __restrict__ ns,
                          float* __restrict__ out, int B)
{
  __shared__ float sf[256];
  __shared__ int   si[256];
  const int t = threadIdx.x;
  float s = 0.0f; int c = 0;
  for (int k = t; k < B; k += 256) { s += partial[k]; c += ns[k]; }
  sf[t] = s; si[t] = c;
  __syncthreads();
  for (int off = 128; off >= 1; off >>= 1) {
    if (t < off) { sf[t] += sf[t + off]; si[t] += si[t + off]; }
    __syncthreads();
  }
  if (t == 0) out[0] = sf[0] / (float)si[0];
}

extern "C" void kernel_launch(void* const* d_in, const int* in_sizes, int n_in,
                              void* d_out, int out_size, void* d_ws, size_t ws_size,
                              hipStream_t stream) {
  const float* pred = (const float*)d_in[0];   // [B,64,64] f32
  const float* gt   = (const float*)d_in[1];   // [B,64,64] f32
  const int*   pns  = (const int*)d_in[2];     // [B] i32 (== gt_ns)
  (void)n_in; (void)out_size; (void)ws_size;

  const int B = in_sizes[2];
  float* partial = (float*)d_ws;               // B floats of scratch

  hung_loss_kernel<<<B, 32, 0, stream>>>(pred, gt, pns, partial, B);
  hung_finalize_kernel<<<1, 256, 0, stream>>>(partial, pns, (float*)d_out, B);
}